// GaussianModel_7791070675873
// MI455X (gfx1250) — compile-verified
//
#include <hip/hip_runtime.h>

// Sigma = R * diag(exp(ls))^2 * R^T per Gaussian (N = 4M).
// Bandwidth-bound: 64 B/elem -> 256 MB stream, ~11 us floor at 23.3 TB/s.
// Data path: b128/b96 nontemporal loads, LDS transpose of the 9-float output
// rows so every global store is a lane-contiguous b128 per wave32, NT store
// hints (single-pass stream, don't rinse L2).
// VALU trimmed with hardware transcendentals (v_rsq_f32, v_exp_f32) so compute
// issue stays well under the memory window (~58% -> ~30% of SIMD slots).

typedef float v4f __attribute__((ext_vector_type(4)));

constexpr int BLOCK = 256;

__global__ __launch_bounds__(BLOCK, 8)
void gs_covariance_kernel(const float* __restrict__ rot,    // (N,4)
                          const float* __restrict__ lsc,    // (N,3)
                          float* __restrict__ out,          // (N,9)
                          int n) {
    __shared__ alignas(16) float tile[BLOCK * 9];   // 9216 B of 320 KB/WGP

    const int  t    = threadIdx.x;
    const long base = (long)blockIdx.x * BLOCK;
    const long i    = base + t;

    if (i < n) {
        // rotation: one b128 NT load, lane-contiguous 16 B -> perfect coalescing
        const v4f q4 = __builtin_nontemporal_load(
            reinterpret_cast<const v4f*>(rot) + i);
        // log_scaling: contiguous 12 B/lane -> single b96 NT load
        const float* lp = lsc + i * 3;
        const float l0 = __builtin_nontemporal_load(lp + 0);
        const float l1 = __builtin_nontemporal_load(lp + 1);
        const float l2 = __builtin_nontemporal_load(lp + 2);

        // normalize quaternion: q * rsqrt(max(|q|^2, 1e-24))  ==
        //                       q / max(|q|, 1e-12)   (reference semantics)
        float w = q4.x, x = q4.y, y = q4.z, z = q4.w;
        const float d   = w * w + x * x + y * y + z * z;
        const float inv = __builtin_amdgcn_rsqf(fmaxf(d, 1e-24f)); // v_rsq_f32
        w *= inv; x *= inv; y *= inv; z *= inv;

        // scales: exp(l) = exp2(l * log2(e)) -> v_exp_f32 (SCALING_MODIFIER=1)
        constexpr float LOG2E = 1.4426950408889634f;
        const float sx = __builtin_amdgcn_exp2f(l0 * LOG2E);
        const float sy = __builtin_amdgcn_exp2f(l1 * LOG2E);
        const float sz = __builtin_amdgcn_exp2f(l2 * LOG2E);

        // rotation matrix
        const float xx = x * x, yy = y * y, zz = z * z;
        const float xy = x * y, xz = x * z, yz = y * z;
        const float wx = w * x, wy = w * y, wz = w * z;

        const float r00 = 1.0f - 2.0f * (yy + zz);
        const float r01 = 2.0f * (xy - wz);
        const float r02 = 2.0f * (xz + wy);
        const float r10 = 2.0f * (xy + wz);
        const float r11 = 1.0f - 2.0f * (xx + zz);
        const float r12 = 2.0f * (yz - wx);
        const float r20 = 2.0f * (xz - wy);
        const float r21 = 2.0f * (yz + wx);
        const float r22 = 1.0f - 2.0f * (xx + yy);

        // M = R * diag(s)
        const float m00 = r00 * sx, m01 = r01 * sy, m02 = r02 * sz;
        const float m10 = r10 * sx, m11 = r11 * sy, m12 = r12 * sz;
        const float m20 = r20 * sx, m21 = r21 * sy, m22 = r22 * sz;

        // Sigma = M * M^T (symmetric; 6 unique dot products)
        const float s00 = m00 * m00 + m01 * m01 + m02 * m02;
        const float s01 = m00 * m10 + m01 * m11 + m02 * m12;
        const float s02 = m00 * m20 + m01 * m21 + m02 * m22;
        const float s11 = m10 * m10 + m11 * m11 + m12 * m12;
        const float s12 = m10 * m20 + m11 * m21 + m12 * m22;
        const float s22 = m20 * m20 + m21 * m21 + m22 * m22;

        // stride-9 dword writes: gcd(9,64 banks)=1 -> conflict-free
        float* r = tile + t * 9;
        r[0] = s00; r[1] = s01; r[2] = s02;
        r[3] = s01; r[4] = s11; r[5] = s12;
        r[6] = s02; r[7] = s12; r[8] = s22;
    }

    __syncthreads();

    // coalesced writeback: BLOCK*9 floats = 576 float4 per full block
    const int m = (int)min((long)BLOCK, (long)n - base);  // valid elems (>=1)
    float* obase = out + base * 9;                        // 9216B-aligned blocks

    if (m == BLOCK) {
        const v4f* t4 = reinterpret_cast<const v4f*>(tile);
        v4f*       o4 = reinterpret_cast<v4f*>(obase);
        __builtin_nontemporal_store(t4[t],           o4 + t);
        __builtin_nontemporal_store(t4[t + BLOCK],   o4 + t + BLOCK);
        if (t < 64)
            __builtin_nontemporal_store(t4[t + 2 * BLOCK], o4 + t + 2 * BLOCK);
    } else {
        const int nf = m * 9;
        #pragma unroll
        for (int k = 0; k < 9; ++k) {
            const int idx = t + BLOCK * k;   // lane-contiguous b32 stores
            if (idx < nf)
                __builtin_nontemporal_store(tile[idx], obase + idx);
        }
    }
}

extern "C" void kernel_launch(void* const* d_in, const int* in_sizes, int n_in,
                              void* d_out, int out_size, void* d_ws, size_t ws_size,
                              hipStream_t stream) {
    const float* rot = (const float*)d_in[0];   // (N,4) float32
    const float* lsc = (const float*)d_in[1];   // (N,3) float32
    float*       out = (float*)d_out;           // (N,3,3) float32

    const int n = in_sizes[0] / 4;              // N elements
    const int grid = (n + BLOCK - 1) / BLOCK;

    gs_covariance_kernel<<<grid, BLOCK, 0, stream>>>(rot, lsc, out, n);
}